// Criterion_44770739093897
// MI455X (gfx1250) — compile-verified
//
#include <hip/hip_runtime.h>
#include <math.h>

// ---------------------------------------------------------------------------
// Hyperbolic triplet criterion, fused pipeline for gfx1250 (MI455X).
// Heavy op: 8192x8192x128 f32 Gram via V_WMMA_F32_16X16X4_F32, fused with
// per-row top-6 selection (D never materialized: 256MB > L2; x = 4MB is
// L2-resident). Column tiles are staged with GLOBAL_LOAD_ASYNC_TO_LDS_B128
// (ASYNCcnt DMA) into double-buffered LDS, overlapping DMA with WMMA.
// ---------------------------------------------------------------------------

#define BN 8192
#define DIM 128
#define NA 32

typedef __attribute__((ext_vector_type(2))) float v2f;
typedef __attribute__((ext_vector_type(8))) float v8f;

// -------------------- Kernel 1: project to Poincare ball --------------------
__global__ void __launch_bounds__(128) k_project(const float* __restrict__ in,
                                                 float* __restrict__ xp,
                                                 float* __restrict__ x2,
                                                 float* __restrict__ w) {
  const int row = blockIdx.x;
  const int t = threadIdx.x;
  float v = in[row * DIM + t];
  __shared__ float red[128];
  red[t] = v * v;
  __syncthreads();
  for (int o = 64; o > 0; o >>= 1) {
    if (t < o) red[t] += red[t + o];
    __syncthreads();
  }
  float s = red[0];
  float norm = fmaxf(sqrtf(s), 1e-15f);
  const float maxn = 0.999f;  // (1 - BALL_EPS)/sqrt(c)
  float scale = (norm > maxn) ? (maxn / norm) : 1.0f;
  xp[row * DIM + t] = v * scale;
  if (t == 0) {
    float xx = s * scale * scale;
    x2[row] = xx;
    w[row] = 1.0f / (1.0f - xx);
  }
}

// -------------------- branchless sorted top-6 insertion ---------------------
// strictly-less insertion + monotonically increasing j  ==> ties keep the
// earliest index, matching jax.lax.top_k tie-break.
__device__ __forceinline__ void insert6(float v, int j, float lv[6], int li[6]) {
  bool c0 = v < lv[0], c1 = v < lv[1], c2 = v < lv[2];
  bool c3 = v < lv[3], c4 = v < lv[4], c5 = v < lv[5];
  lv[5] = c4 ? lv[4] : (c5 ? v : lv[5]); li[5] = c4 ? li[4] : (c5 ? j : li[5]);
  lv[4] = c3 ? lv[3] : (c4 ? v : lv[4]); li[4] = c3 ? li[3] : (c4 ? j : li[4]);
  lv[3] = c2 ? lv[2] : (c3 ? v : lv[3]); li[3] = c2 ? li[2] : (c3 ? j : li[3]);
  lv[2] = c1 ? lv[1] : (c2 ? v : lv[2]); li[2] = c1 ? li[1] : (c2 ? j : li[2]);
  lv[1] = c0 ? lv[0] : (c1 ? v : lv[1]); li[1] = c0 ? li[0] : (c1 ? j : li[1]);
  lv[0] = c0 ? v : lv[0];                li[0] = c0 ? j : li[0];
}

// ---- Kernel 2: fused WMMA distance + per-row top-6 -> connectivity count ---
#define XSTRIDE 132  // 32 column rows of 128 floats, padded (16B align + banks)

// Async DMA of one 32-point tile (32x128 f32 = 16KB) into LDS, 256 threads,
// 4x GLOBAL_LOAD_ASYNC_TO_LDS_B128 per thread. Completion via ASYNCcnt.
__device__ __forceinline__ void stage_tile_async(const float* __restrict__ xp,
                                                 int colbase, float* buf,
                                                 int tid) {
#pragma unroll
  for (int q = 0; q < 4; q++) {
    int f = (q * 256 + tid) * 4;
    int c = f >> 7, k = f & 127;
    const float* g = &xp[(colbase + c) * DIM + k];
    // LDS aperture keeps the local byte offset in the low 32 bits (ISA 10.2)
    unsigned ldsoff = (unsigned)(size_t)&buf[c * XSTRIDE + k];
    asm volatile("global_load_async_to_lds_b128 %0, %1, off"
                 :
                 : "v"(ldsoff), "v"(g)
                 : "memory");
  }
}

__global__ void __launch_bounds__(256) k_disttop(const float* __restrict__ xp,
                                                 const float* __restrict__ x2,
                                                 const float* __restrict__ w,
                                                 const int* __restrict__ labels,
                                                 int* __restrict__ conn) {
  __shared__ float ldsX[2][32 * XSTRIDE];  // double-buffered column tiles
  __shared__ float xch[8][16 * 36];        // per-wave 16x32 tile, row stride 36
  __shared__ float mgV[8][16 * 12];
  __shared__ int   mgI[8][16 * 12];
  __shared__ float sX2[BN];                // whole x2 / w arrays, LDS-resident
  __shared__ float sW[BN];

  const int tid = threadIdx.x;
  const int wv = tid >> 5, lane = tid & 31;
  const int hi = lane >> 4, ln = lane & 15;
  const int R = (blockIdx.x * 8 + wv) * 16;  // 16-row tile owned by this wave

  // stage x2/w once (32KB each)
  for (int i = tid; i < BN; i += 256) {
    sX2[i] = x2[i];
    sW[i] = w[i];
  }

  // A tile: 16 rows x 128 dims. f32 16x4 A layout: lanes 0-15 = M,
  // VGPR0 = {K, K+2}(lo/hi half), VGPR1 = {K+1, K+3}  -> contiguous float2.
  v2f a[32];
#pragma unroll
  for (int kc = 0; kc < 32; kc++)
    a[kc] = *(const v2f*)&xp[(R + ln) * DIM + kc * 4 + 2 * hi];

  // async prologue: DMA tile 0 into buffer 0
  stage_tile_async(xp, 0, &ldsX[0][0], tid);

  __syncthreads();  // sX2/sW ready

  float x2i[8];
#pragma unroll
  for (int r = 0; r < 8; r++) x2i[r] = sX2[R + r + 8 * hi];

  float lv[6]; int li[6];
#pragma unroll
  for (int k = 0; k < 6; k++) { lv[k] = 3.0e38f; li[k] = 0x7fffffff; }

  for (int cb = 0; cb < BN / 32; ++cb) {
    const int colbase = cb * 32;
    float* cbuf = &ldsX[cb & 1][0];
    float* nbuf = &ldsX[(cb & 1) ^ 1][0];

    // wait for this wave's DMA of the current tile; barrier joins all waves
    asm volatile("s_wait_asynccnt 0x0" ::: "memory");
    __syncthreads();

    // kick off DMA for the next tile; overlaps with the WMMA work below
    if (cb + 1 < BN / 32) stage_tile_async(xp, colbase + 32, nbuf, tid);

    const float x2j0 = sX2[colbase + ln],      wj0 = sW[colbase + ln];
    const float x2j1 = sX2[colbase + 16 + ln], wj1 = sW[colbase + 16 + ln];

    v8f c0 = {0, 0, 0, 0, 0, 0, 0, 0};
    v8f c1 = {0, 0, 0, 0, 0, 0, 0, 0};
#pragma unroll
    for (int kc = 0; kc < 32; kc++) {
      v2f b0 = *(const v2f*)&cbuf[ln * XSTRIDE + kc * 4 + 2 * hi];
      v2f b1 = *(const v2f*)&cbuf[(16 + ln) * XSTRIDE + kc * 4 + 2 * hi];
      c0 = __builtin_amdgcn_wmma_f32_16x16x4_f32(false, a[kc], false, b0,
                                                 (short)0, c0, false, false);
      c1 = __builtin_amdgcn_wmma_f32_16x16x4_f32(false, a[kc], false, b1,
                                                 (short)0, c1, false, false);
    }

    // monotone ranking key: t = max(x2_i + x2_j - 2*dot, 0) * w_j
    // (w_i is a positive per-row constant -> same ordering as arccosh dist).
    // C layout: VGPR r holds row M = r + 8*hi, column N = ln.
#pragma unroll
    for (int r = 0; r < 8; r++) {
      int M = r + 8 * hi;
      float t0 = fmaxf(x2i[r] + x2j0 - 2.0f * c0[r], 0.0f) * wj0;
      float t1 = fmaxf(x2i[r] + x2j1 - 2.0f * c1[r], 0.0f) * wj1;
      xch[wv][M * 36 + ln] = t0;
      xch[wv][M * 36 + 16 + ln] = t1;
    }
    __syncthreads();

    // transpose consumed: lane owns row ln, columns hi*16 .. hi*16+15
#pragma unroll
    for (int c = 0; c < 16; c++) {
      int cc = hi * 16 + c;
      insert6(xch[wv][ln * 36 + cc], colbase + cc, lv, li);
    }
    // next iteration's leading barrier orders these reads vs next xch writes
  }
  __syncthreads();

  // merge the two half-lists per row (lanes L and L+16 both handled row L)
#pragma unroll
  for (int k = 0; k < 6; k++) {
    mgV[wv][ln * 12 + hi * 6 + k] = lv[k];
    mgI[wv][ln * 12 + hi * 6 + k] = li[k];
  }
  __syncthreads();
  if (hi == 0) {
    const float* va = &mgV[wv][ln * 12]; const int* ia = &mgI[wv][ln * 12];
    const float* vb = va + 6;            const int* ib = ia + 6;
    int myLab = labels[R + ln];
    int p = 0, q = 0, cnt = 0;
    for (int k = 0; k < 6; k++) {
      bool takeA = (p < 6) && (q >= 6 || (va[p] < vb[q]) ||
                               (va[p] == vb[q] && ia[p] < ib[q]));
      int j;
      if (takeA) { j = ia[p]; p++; } else { j = ib[q]; q++; }
      if (k > 0) cnt += (labels[j] == myLab) ? 1 : 0;  // rank 0 == self
    }
    conn[R + ln] = cnt;  // connectivity = cnt/5 (monotone in cnt)
  }
}

// ---- Kernel 3: top-32 of connectivity with exact jax top_k tie-break -------
__global__ void __launch_bounds__(256) k_anchors(const int* __restrict__ conn,
                                                 const int* __restrict__ labels,
                                                 int* __restrict__ anchors,
                                                 int* __restrict__ aLab) {
  __shared__ int cnt[6];
  __shared__ int take[6];
  __shared__ int baseA[6];
  __shared__ int pfx[257];
  const int tid = threadIdx.x;
  if (tid < 6) cnt[tid] = 0;
  __syncthreads();
  for (int i = tid; i < BN; i += 256) atomicAdd(&cnt[conn[i]], 1);
  __syncthreads();
  if (tid == 0) {
    int rem = NA, base = 0;
    for (int b = 5; b >= 0; b--) {
      int t = min(cnt[b], rem);
      take[b] = t; baseA[b] = base; base += t; rem -= t;
    }
  }
  __syncthreads();
  for (int b = 5; b >= 0; b--) {
    int t = take[b];  // uniform
    if (t > 0) {
      int local = 0;
      int s = tid * 32;
      for (int k = 0; k < 32; k++) local += (conn[s + k] == b) ? 1 : 0;
      pfx[tid + 1] = local;
      if (tid == 0) pfx[0] = 0;
      __syncthreads();
      if (tid == 0)
        for (int k = 1; k <= 256; k++) pfx[k] += pfx[k - 1];
      __syncthreads();
      int pos = baseA[b] + pfx[tid];
      int lim = baseA[b] + t;
      for (int k = 0; k < 32; k++) {
        if (conn[s + k] == b) {
          if (pos < lim) anchors[pos] = s + k;
          pos++;
        }
      }
      __syncthreads();
    }
  }
  __syncthreads();
  if (tid < NA) aLab[tid] = labels[anchors[tid]];
}

// -------------------- Kernel 4: Da = D[anchors]  (32 x 8192) ----------------
__global__ void __launch_bounds__(256) k_da(const float* __restrict__ xp,
                                            const float* __restrict__ x2,
                                            const float* __restrict__ w,
                                            const int* __restrict__ anchors,
                                            float* __restrict__ Da) {
  __shared__ float xa[DIM];
  __shared__ float sc[2];
  const int a = blockIdx.y;
  const int tid = threadIdx.x;
  int anc = anchors[a];
  if (tid < DIM) xa[tid] = xp[anc * DIM + tid];
  if (tid == 128) { sc[0] = x2[anc]; sc[1] = w[anc]; }
  __syncthreads();
  const int j = blockIdx.x * 256 + tid;
  float dot = 0.0f;
#pragma unroll 8
  for (int k = 0; k < DIM; k++) dot += xa[k] * xp[j * DIM + k];
  float sq = fmaxf(sc[0] + x2[j] - 2.0f * dot, 0.0f);
  float arg = fmaxf(1.0f + 2.0f * sq * sc[1] * w[j], 1.0f + 1e-7f);
  Da[a * BN + j] = acoshf(arg);
}

// -------------------- threefry2x32 (20 rounds, JAX-exact) -------------------
__device__ __forceinline__ unsigned rotl32(unsigned x, int n) {
  return (x << n) | (x >> (32 - n));
}
__device__ __forceinline__ void threefry(unsigned k0, unsigned k1, unsigned x0,
                                         unsigned x1, unsigned& o0,
                                         unsigned& o1) {
  unsigned ks2 = k0 ^ k1 ^ 0x1BD11BDAu;
  unsigned v0 = x0 + k0, v1 = x1 + k1;
  const int r1[4] = {13, 15, 26, 6};
  const int r2[4] = {17, 29, 16, 24};
#pragma unroll
  for (int i = 0; i < 4; i++) { v0 += v1; v1 = rotl32(v1, r1[i]); v1 ^= v0; }
  v0 += k1; v1 += ks2 + 1u;
#pragma unroll
  for (int i = 0; i < 4; i++) { v0 += v1; v1 = rotl32(v1, r2[i]); v1 ^= v0; }
  v0 += ks2; v1 += k0 + 2u;
#pragma unroll
  for (int i = 0; i < 4; i++) { v0 += v1; v1 = rotl32(v1, r1[i]); v1 ^= v0; }
  v0 += k0; v1 += k1 + 3u;
#pragma unroll
  for (int i = 0; i < 4; i++) { v0 += v1; v1 = rotl32(v1, r2[i]); v1 ^= v0; }
  v0 += k1; v1 += ks2 + 4u;
#pragma unroll
  for (int i = 0; i < 4; i++) { v0 += v1; v1 = rotl32(v1, r1[i]); v1 ^= v0; }
  v0 += ks2; v1 += k0 + 5u;
  o0 = v0; o1 = v1;
}

// ------- Kernel 5: jax.random.categorical (Gumbel argmax, bit-exact) --------
__global__ void __launch_bounds__(256) k_sample(const float* __restrict__ Da,
                                                const int* __restrict__ labels,
                                                const int* __restrict__ anchors,
                                                const int* __restrict__ aLab,
                                                int* __restrict__ outIdx) {
  const int row = blockIdx.x & 31;    // anchor
  const int isNeg = blockIdx.x >> 5;  // 0 = pos (kp), 1 = neg (kn)
  const int tid = threadIdx.x;
  // jax.random.key(42) = (0,42); split: threefry over counts [0,1,2,3]
  unsigned a0, b0, a1, b1;
  threefry(0u, 42u, 0u, 2u, a0, b0);
  threefry(0u, 42u, 1u, 3u, a1, b1);
  unsigned k0 = isNeg ? b0 : a0;
  unsigned k1 = isNeg ? b1 : a1;

  const int anc = anchors[row], al = aLab[row];
  const float TINY = 1.17549435e-38f;
  float best = -3.4e38f; int bestJ = 0x7fffffff;
  for (int j = tid; j < BN; j += 256) {
    int lab = labels[j];
    float d = Da[row * BN + j];
    float logit;
    if (!isNeg) logit = ((lab == al) && (j != anc)) ? (-d * 10.0f) : -1e30f;
    else        logit = (lab != al) ? (d * 10.0f) : -1e30f;
    unsigned i = (unsigned)(row * BN + j);
    unsigned x0, x1; int sel;
    if (i < 131072u) { x0 = i; x1 = i + 131072u; sel = 0; }
    else             { x0 = i - 131072u; x1 = i; sel = 1; }
    unsigned o0, o1;
    threefry(k0, k1, x0, x1, o0, o1);
    unsigned bits = sel ? o1 : o0;
    float u = __uint_as_float((bits >> 9) | 0x3f800000u) - 1.0f;
    float uv = fmaxf(TINY, u * (1.0f - TINY) + TINY);
    float g = -logf(-logf(uv));
    float s = logit + g;
    if ((s > best) || (s == best && j < bestJ)) { best = s; bestJ = j; }
  }
  __shared__ float sv[256];
  __shared__ int si[256];
  sv[tid] = best; si[tid] = bestJ;
  __syncthreads();
  for (int o = 128; o > 0; o >>= 1) {
    if (tid < o) {
      if ((sv[tid + o] > sv[tid]) ||
          (sv[tid + o] == sv[tid] && si[tid + o] < si[tid])) {
        sv[tid] = sv[tid + o]; si[tid] = si[tid + o];
      }
    }
    __syncthreads();
  }
  if (tid == 0) outIdx[isNeg * NA + row] = si[0];
}

// -------------------- Kernel 6: triplet loss mean ---------------------------
__global__ void k_loss(const float* __restrict__ xp, const float* __restrict__ x2,
                       const int* __restrict__ anchors,
                       const int* __restrict__ idxPN, float* __restrict__ out) {
  const int a = threadIdx.x;  // 32 threads
  __shared__ float red[NA];
  int ia = anchors[a], ip = idxPN[a], inn = idxPN[NA + a];
  float sqp = 0.0f, sqn = 0.0f;
  for (int k = 0; k < DIM; k++) {
    float va = xp[ia * DIM + k];
    float d1 = va - xp[ip * DIM + k];  sqp += d1 * d1;
    float d2 = va - xp[inn * DIM + k]; sqn += d2 * d2;
  }
  float wa = 1.0f / (1.0f - x2[ia]);
  float dp = acoshf(fmaxf(1.0f + 2.0f * sqp * wa / (1.0f - x2[ip]), 1.0f + 1e-7f));
  float dn = acoshf(fmaxf(1.0f + 2.0f * sqn * wa / (1.0f - x2[inn]), 1.0f + 1e-7f));
  red[a] = fmaxf(dp - dn + 0.2f, 0.0f);
  __syncthreads();
  if (a == 0) {
    float s = 0.0f;
    for (int k = 0; k < NA; k++) s += red[k];
    out[0] = s / (float)NA;
  }
}

// ---------------------------------------------------------------------------
extern "C" void kernel_launch(void* const* d_in, const int* in_sizes, int n_in,
                              void* d_out, int out_size, void* d_ws,
                              size_t ws_size, hipStream_t stream) {
  const float* batch = (const float*)d_in[0];
  const int* labels = (const int*)d_in[1];
  float* out = (float*)d_out;

  float* xp = (float*)d_ws;            // 8192*128
  float* x2 = xp + BN * DIM;           // 8192
  float* w = x2 + BN;                  // 8192
  float* Da = w + BN;                  // 32*8192
  int* conn = (int*)(Da + NA * BN);    // 8192
  int* anchors = conn + BN;            // 32
  int* aLab = anchors + NA;            // 32
  int* idxPN = aLab + NA;              // 64

  k_project<<<BN, 128, 0, stream>>>(batch, xp, x2, w);
  k_disttop<<<BN / 128, 256, 0, stream>>>(xp, x2, w, labels, conn);
  k_anchors<<<1, 256, 0, stream>>>(conn, labels, anchors, aLab);
  k_da<<<dim3(BN / 256, NA), 256, 0, stream>>>(xp, x2, w, anchors, Da);
  k_sample<<<2 * NA, 256, 0, stream>>>(Da, labels, anchors, aLab, idxPN);
  k_loss<<<1, NA, 0, stream>>>(xp, x2, anchors, idxPN, out);
}